// CommAwareRGAT_53025666237105
// MI455X (gfx1250) — compile-verified
//
#include <hip/hip_runtime.h>
#include <math.h>

#define NN 100000
#define EE 500000
#define H  128
#define OD 64
#define LAY 2
#define REL 2

typedef float v2f __attribute__((ext_vector_type(2)));
typedef float v8f __attribute__((ext_vector_type(8)));
typedef int   v4i __attribute__((vector_size(16)));          // GCC-vector int4
typedef __attribute__((address_space(1))) v4i* as1_v4i;      // global
typedef __attribute__((address_space(3))) v4i* as3_v4i;      // LDS

#if defined(__HIP_DEVICE_COMPILE__) && __has_builtin(__builtin_amdgcn_global_load_async_to_lds_b128)
#define HAVE_ASYNC_LDS 1
#else
#define HAVE_ASYNC_LDS 0
#endif

__device__ __forceinline__ void wait_async_lds() {
#if HAVE_ASYNC_LDS
#if __has_builtin(__builtin_amdgcn_s_wait_asynccnt)
  __builtin_amdgcn_s_wait_asynccnt(0);
#else
  asm volatile("s_wait_asynccnt 0x0" ::: "memory");
#endif
#endif
}

// ---------------------------------------------------------------------------
// WMMA f32 GEMM: C[M x NO] = A[M x 128] @ W[128 x NO] (+bias) (+beta*C)
// Workgroup = 4 waves = 64 rows x 64 cols. The 128x16 W slice (cols n0..n0+15) is staged into
// LDS once per workgroup via async-to-LDS (ASYNCcnt-tracked, bypasses the
// wave's load pipe), then each wave computes a 16x64 tile with
// V_WMMA_F32_16X16X4_F32, reading B fragments from LDS (ds_load).
// A-fragment (16x4 f32): lanes 0-15 hold M=lane, K=k+0/k+1; lanes 16-31 hold
// M=lane-16, K=k+2/k+3. C/D: reg i -> row m0+i (lanes 0-15) / m0+i+8 (16-31).
// ---------------------------------------------------------------------------
#define LWS 68  // LDS row stride for W panel (64 cols + 4 pad floats)

template <int BETA, int BIAS>
__global__ void gemm_wmma(const float* __restrict__ A, const float* __restrict__ W,
                          const float* __restrict__ bias, float* __restrict__ C,
                          int M, int NO) {
  __shared__ float lW[128 * LWS];                  // 34 KB
  const int lane = threadIdx.x & 31;
  const int n0   = blockIdx.y * 64;
  const int tid  = threadIdx.y * 32 + threadIdx.x; // 0..127

  // ---- cooperative stage of W[0:128, n0:n0+16] into LDS (16B chunks) ----
  // 128 rows * 4 chunks = 512 chunks; 128 threads -> exactly 4 each (no div.)
#pragma unroll
  for (int it = 0; it < 4; ++it) {
    const int c   = tid + it * 128;
    const int row = c >> 2;
    const int ch  = (c & 3) * 4;
    const float* g = W + (size_t)row * NO + n0 + ch;
    float*       l = lW + row * LWS + ch;
#if HAVE_ASYNC_LDS
    __builtin_amdgcn_global_load_async_to_lds_b128((as1_v4i)g, (as3_v4i)l, 0, 0);
#else
    *(float4*)l = *(const float4*)g;
#endif
  }
  wait_async_lds();
  __syncthreads();

  const int tm = blockIdx.x * 4 + threadIdx.y;     // 16-row tile index
  const int m0 = tm * 16;
  if (m0 >= M) return;                             // wave-uniform (after barrier)
  const int half = lane >> 4;                      // K sub-step select
  const int lid  = lane & 15;                      // row / col within tile
  const float* Arow = A + (size_t)(m0 + lid) * H;

  v8f acc0 = {}, acc1 = {}, acc2 = {}, acc3 = {};
  for (int k = 0; k < H; k += 4) {
    if (k + 32 < H) __builtin_prefetch(Arow + k + 32, 0, 0);
    v2f a;
    a.x = Arow[k + half * 2 + 0];
    a.y = Arow[k + half * 2 + 1];
    const float* Lr0 = lW + (k + half * 2 + 0) * LWS + lid;
    const float* Lr1 = lW + (k + half * 2 + 1) * LWS + lid;
    v2f b0, b1, b2, b3;
    b0.x = Lr0[0];  b0.y = Lr1[0];
    b1.x = Lr0[16]; b1.y = Lr1[16];
    b2.x = Lr0[32]; b2.y = Lr1[32];
    b3.x = Lr0[48]; b3.y = Lr1[48];
    acc0 = __builtin_amdgcn_wmma_f32_16x16x4_f32(false, a, false, b0, (short)0, acc0, false, false);
    acc1 = __builtin_amdgcn_wmma_f32_16x16x4_f32(false, a, false, b1, (short)0, acc1, false, false);
    acc2 = __builtin_amdgcn_wmma_f32_16x16x4_f32(false, a, false, b2, (short)0, acc2, false, false);
    acc3 = __builtin_amdgcn_wmma_f32_16x16x4_f32(false, a, false, b3, (short)0, acc3, false, false);
  }
  v8f accs[4] = {acc0, acc1, acc2, acc3};
#pragma unroll
  for (int t = 0; t < 4; ++t) {
    const int col = n0 + t * 16 + lid;
    const float bv = BIAS ? bias[col] : 0.0f;
#pragma unroll
    for (int i = 0; i < 8; ++i) {
      const int r = m0 + i + half * 8;
      size_t idx = (size_t)r * NO + col;
      float v = accs[t][i] + bv;
      if (BETA) v += C[idx];
      C[idx] = v;
    }
  }
}

// out[n] = dot(X[n, 0:128], w[0:128])  -- one wave per row, float4 lanes
__global__ void rowdot128(const float* __restrict__ X, const float* __restrict__ w,
                          float* __restrict__ out, int nrows) {
  const int wid  = blockIdx.x * (blockDim.x >> 5) + (threadIdx.x >> 5);
  const int lane = threadIdx.x & 31;
  if (wid >= nrows) return;
  const float4 x = ((const float4*)(X + (size_t)wid * H))[lane];
  const float4 c = ((const float4*)w)[lane];
  float s = x.x * c.x + x.y * c.y + x.z * c.z + x.w * c.w;
#pragma unroll
  for (int off = 16; off > 0; off >>= 1) s += __shfl_down(s, off, 32);
  if (lane == 0) out[wid] = s;
}

__global__ void zero_f32(float* __restrict__ p, int n) {
  int i = blockIdx.x * blockDim.x + threadIdx.x;
  if (i < n) p[i] = 0.0f;
}

// per-edge score: s = leakyrelu(al[dst]+ar[src]+pb); num=exp(s); denom[dst]+=num
__global__ void edge_score(const int* __restrict__ src, const int* __restrict__ dst,
                           const float* __restrict__ al, const float* __restrict__ ar,
                           const float* __restrict__ pb, float* __restrict__ num,
                           float* __restrict__ denom, int ne) {
  int e = blockIdx.x * blockDim.x + threadIdx.x;
  if (e >= ne) return;
  int s = src[e], d = dst[e];
  float sc = al[d] + ar[s] + pb[0];
  sc = sc > 0.0f ? sc : 0.2f * sc;
  float nm = expf(sc);
  num[e] = nm;
  unsafeAtomicAdd(&denom[d], nm);
}

// per-edge aggregation: out[dst,:] += hj[src,:] * num[e]/(denom[src]+1e-16)
// (denominator gathered at SRC, faithful to the reference.)  One wave per edge.
__global__ void edge_aggr(const int* __restrict__ src, const int* __restrict__ dst,
                          const float* __restrict__ num, const float* __restrict__ denom,
                          const float* __restrict__ hj, float* __restrict__ out, int ne) {
  const int e    = blockIdx.x * (blockDim.x >> 5) + (threadIdx.x >> 5);
  const int lane = threadIdx.x & 31;
  if (e >= ne) return;
  const int s = src[e], d = dst[e];
  const float alpha = num[e] / (denom[s] + 1e-16f);
  const float4 hv = ((const float4*)(hj + (size_t)s * H))[lane];
  float* op = out + (size_t)d * H + lane * 4;
  unsafeAtomicAdd(op + 0, hv.x * alpha);
  unsafeAtomicAdd(op + 1, hv.y * alpha);
  unsafeAtomicAdd(op + 2, hv.z * alpha);
  unsafeAtomicAdd(op + 3, hv.w * alpha);
}

// Column stats over node dim: mu[c], rsig[c] = rsqrt(var+1e-5). 1 block / col.
__global__ void bn_stats(const float* __restrict__ X, int nrows,
                         float* __restrict__ mu, float* __restrict__ rsig) {
  const int c = blockIdx.x;
  float s = 0.0f, s2 = 0.0f;
  for (int r = threadIdx.x; r < nrows; r += blockDim.x) {
    float v = X[(size_t)r * H + c];
    s += v; s2 += v * v;
  }
  __shared__ float sh[256], sh2[256];
  sh[threadIdx.x] = s; sh2[threadIdx.x] = s2;
  __syncthreads();
  for (int o = blockDim.x >> 1; o > 0; o >>= 1) {
    if ((int)threadIdx.x < o) { sh[threadIdx.x] += sh[threadIdx.x + o]; sh2[threadIdx.x] += sh2[threadIdx.x + o]; }
    __syncthreads();
  }
  if (threadIdx.x == 0) {
    float m = sh[0] / (float)nrows;
    float var = sh2[0] / (float)nrows - m * m;
    mu[c] = m;
    rsig[c] = rsqrtf(var + 1e-5f);
  }
}

// y = relu((x-mu)*rsig*g + b)
__global__ void bn_relu(const float* __restrict__ X, const float* __restrict__ mu,
                        const float* __restrict__ rsig, const float* __restrict__ g,
                        const float* __restrict__ b, float* __restrict__ Y, size_t n) {
  size_t i = (size_t)blockIdx.x * blockDim.x + threadIdx.x;
  if (i >= n) return;
  int c = (int)(i & (H - 1));
  float v = (X[i] - mu[c]) * rsig[c] * g[c] + b[c];
  Y[i] = v > 0.0f ? v : 0.0f;
}

// ---------------------------------------------------------------------------
extern "C" void kernel_launch(void* const* d_in, const int* in_sizes, int n_in,
                              void* d_out, int out_size, void* d_ws, size_t ws_size,
                              hipStream_t stream) {
  const float* x0       = (const float*)d_in[0];
  const float* x1       = (const float*)d_in[1];
  const int*   e0       = (const int*)d_in[2];
  const int*   e1       = (const int*)d_in[3];
  const float* conv_w   = (const float*)d_in[4];
  const float* proj_w   = (const float*)d_in[5];
  const float* proj_b   = (const float*)d_in[6];
  const float* res_w    = (const float*)d_in[7];
  const float* gat_bias = (const float*)d_in[8];
  const float* skip_w   = (const float*)d_in[9];
  const float* skip_b   = (const float*)d_in[10];
  const float* bn_g     = (const float*)d_in[11];
  const float* bn_b     = (const float*)d_in[12];
  const float* mlp_w1   = (const float*)d_in[13];
  const float* mlp_b1   = (const float*)d_in[14];
  const float* mlp_bn_g = (const float*)d_in[15];
  const float* mlp_bn_b = (const float*)d_in[16];
  const float* mlp_w2   = (const float*)d_in[17];
  const float* mlp_b2   = (const float*)d_in[18];

  const size_t NF = (size_t)NN * H;
  float* ws    = (float*)d_ws;
  float* buf0  = ws;              // per-layer output, type 0
  float* buf1  = buf0 + NF;       // per-layer output, type 1
  float* tmp0  = buf1 + NF;       // accumulator, type 0
  float* tmp1  = tmp0 + NF;       // accumulator, type 1
  float* hbuf  = tmp1 + NF;       // h  = x_dst @ cw
  float* hjbuf = hbuf + NF;       // hj = x_src @ cw
  float* al    = hjbuf + NF;      // [NN]
  float* ar    = al + NN;         // [NN]
  float* numv  = ar + NN;         // [EE]
  float* denom = numv + EE;       // [NN]
  float* mu    = denom + NN;      // [H]
  float* rsig  = mu + H;          // [H]

  const dim3 gblk(32, 4);                       // 4 waves / workgroup
  const int  mtiles  = NN / 16;                 // 6250
  const int  gemm_gx = (mtiles + 3) / 4;        // 1563
  const dim3 grd128(gemm_gx, H / 64);           // NO = 128
  const dim3 grd64(gemm_gx, OD / 64);           // NO = 64
  const int  rdg = (NN + 7) / 8;                // rowdot: 8 waves / block
  const int  zg  = (NN + 255) / 256;
  const int  esg = (EE + 255) / 256;
  const int  eag = (EE + 7) / 8;                // edge_aggr: 8 waves / block
  const int  brg = (int)((NF + 255) / 256);

  const float* cur0 = x0;
  const float* cur1 = x1;

  for (int i = 0; i < LAY; ++i) {
    // temp = cur @ skip_w[i] + skip_b[i]
    gemm_wmma<0, 1><<<grd128, gblk, 0, stream>>>(cur0, skip_w + (size_t)i * H * H,
                                                 skip_b + (size_t)i * H, tmp0, NN, H);
    gemm_wmma<0, 1><<<grd128, gblk, 0, stream>>>(cur1, skip_w + (size_t)i * H * H,
                                                 skip_b + (size_t)i * H, tmp1, NN, H);
    for (int j = 0; j < REL; ++j) {
      const int st = (j == 0) ? 0 : 1;
      const int dt = 1 - st;
      const float* xs   = (st == 0) ? cur0 : cur1;
      const float* xd   = (dt == 0) ? cur0 : cur1;
      float*       tdst = (dt == 0) ? tmp0 : tmp1;
      const int*   ed   = (j == 0) ? e0 : e1;
      const int*   srcI = ed;            // e[0]
      const int*   dstI = ed + EE;       // e[1]
      const size_t rij  = (size_t)(i * REL + j);
      const float* cw   = conv_w + rij * H * H;
      const float* pwA  = proj_w + rij * 2 * H;      // rows 0..127
      const float* pwB  = pwA + H;                   // rows 128..255
      const float* pbp  = proj_b + rij;
      const float* rw   = res_w + rij * H * H;
      const float* gb   = gat_bias + rij * H;

      gemm_wmma<0, 0><<<grd128, gblk, 0, stream>>>(xd, cw, nullptr, hbuf, NN, H);
      gemm_wmma<0, 0><<<grd128, gblk, 0, stream>>>(xs, cw, nullptr, hjbuf, NN, H);
      rowdot128<<<rdg, 256, 0, stream>>>(hbuf, pwA, al, NN);
      rowdot128<<<rdg, 256, 0, stream>>>(hjbuf, pwB, ar, NN);
      zero_f32<<<zg, 256, 0, stream>>>(denom, NN);
      edge_score<<<esg, 256, 0, stream>>>(srcI, dstI, al, ar, pbp, numv, denom, EE);
      edge_aggr<<<eag, 256, 0, stream>>>(srcI, dstI, numv, denom, hjbuf, tdst, EE);
      gemm_wmma<1, 1><<<grd128, gblk, 0, stream>>>(xd, rw, gb, tdst, NN, H);
    }
    // BN + ReLU per node type
    bn_stats<<<H, 256, 0, stream>>>(tmp0, NN, mu, rsig);
    bn_relu<<<brg, 256, 0, stream>>>(tmp0, mu, rsig, bn_g + (size_t)i * H,
                                     bn_b + (size_t)i * H, buf0, NF);
    bn_stats<<<H, 256, 0, stream>>>(tmp1, NN, mu, rsig);
    bn_relu<<<brg, 256, 0, stream>>>(tmp1, mu, rsig, bn_g + (size_t)i * H,
                                     bn_b + (size_t)i * H, buf1, NF);
    cur0 = buf0;
    cur1 = buf1;
  }

  // Final MLP on node type 0
  gemm_wmma<0, 1><<<grd128, gblk, 0, stream>>>(cur0, mlp_w1, mlp_b1, hbuf, NN, H);
  bn_stats<<<H, 256, 0, stream>>>(hbuf, NN, mu, rsig);
  bn_relu<<<brg, 256, 0, stream>>>(hbuf, mu, rsig, mlp_bn_g, mlp_bn_b, hjbuf, NF);
  gemm_wmma<0, 1><<<grd64, gblk, 0, stream>>>(hjbuf, mlp_w2, mlp_b2, (float*)d_out, NN, OD);
}